// _HEALPixPadFunction_39178691674830
// MI455X (gfx1250) — compile-verified
//
#include <hip/hip_runtime.h>
#include <stdint.h>

// HEALPix padding for x:[1,12,C,256,256] f32, pad=1 (DLWP-HPX convention).
// Pure data movement: ~811 MB traffic -> HBM-bound (~35us floor @ 23.3 TB/s).
// CDNA5 path: bulk interior copy is fully async through LDS:
//   global_load_async_to_lds_b128  (global -> LDS, 512B/wave/instr, ASYNCcnt)
//   s_wait_asynccnt 0 + s_barrier
//   global_store_async_from_lds_b32 (LDS -> global, alignment-proof for the
//                                    258-stride +1-column output, ASYNCcnt)
// No VGPR staging of the payload at all.

#define HP_H 256
#define HP_W 256
#define HP_F 12
#define HP_ROWS 8   // rows staged per block (8 KB LDS)

// ---------------------------------------------------------------------------
// Interior: out[f,c, y+p, x+p] = in[f,c, y, x], async in both directions.
// grid.x = 12*C*(256/HP_ROWS), block = 256 threads (8 waves, wave32).
// ---------------------------------------------------------------------------
__global__ void healpix_interior_async(const float* __restrict__ in,
                                       float* __restrict__ out,
                                       int C, int HO, int p) {
    __shared__ float tile[HP_ROWS * HP_W];

    const int nrg = HP_H / HP_ROWS;            // 32 row-groups per face-channel
    const int blk = blockIdx.x;
    const int rg  = blk % nrg;
    const int fc  = blk / nrg;                 // face*C + channel
    const int y0  = rg * HP_ROWS;
    const int t   = threadIdx.x;               // 0..255

    const float* src = in  + (size_t)fc * HP_H * HP_W + (size_t)y0 * HP_W;
    float*       dst = out + (size_t)fc * HO * HO + (size_t)(y0 + p) * HO + p;

    // Generic pointer to a __shared__ object: low 32 bits are the LDS byte offset
    // (LDS aperture base occupies addr[63:32] per CDNA5 ISA 10.2).
    const uint32_t ldsBase = (uint32_t)(uintptr_t)(void*)tile;

    // Load leg: 8 rows * 256 floats = 512 x b128 transfers; 2 per thread.
    // Input rows are 1024B-aligned so every b128 load is naturally aligned.
#pragma unroll
    for (int it = 0; it < 2; ++it) {
        const int idx = t + it * 256;                          // 0..511
        const uint64_t g = (uint64_t)(uintptr_t)(src + idx * 4);
        const uint32_t l = ldsBase + (uint32_t)idx * 16u;
        asm volatile("global_load_async_to_lds_b128 %0, %1, off"
                     :: "v"(l), "v"(g)
                     : "memory");
    }
    asm volatile("s_wait_asynccnt 0x0" ::: "memory");
    __syncthreads();   // cross-wave visibility of the staged tile

    // Store leg: thread t emits column t of each row, b32 per lane (128B/wave),
    // straight from LDS through the async engine -- no VGPR payload staging.
#pragma unroll
    for (int r = 0; r < HP_ROWS; ++r) {
        const uint64_t g = (uint64_t)(uintptr_t)(dst + (size_t)r * HO + t);
        const uint32_t l = ldsBase + (uint32_t)(r * HP_W + t) * 4u;
        asm volatile("global_store_async_from_lds_b32 %0, %1, off"
                     :: "v"(g), "v"(l)
                     : "memory");
    }
    // Drain before the workgroup (and its LDS allocation) retires.
    asm volatile("s_wait_asynccnt 0x0" ::: "memory");
}

// ---------------------------------------------------------------------------
// Halo fill for p=1. grid.x = 12*C, block = 256 threads; thread k fills element
// k of all four edges, thread 0 writes the 4 corners. Mappings derived from the
// reference's rot90 algebra, specialized to p=1.
// ---------------------------------------------------------------------------
__global__ void healpix_halo_p1(const float* __restrict__ in,
                                float* __restrict__ out,
                                int C, int HO) {
    const size_t fstride = (size_t)HP_H * HP_W;
    const int fc = blockIdx.x;
    const int f  = fc / C;
    const int c  = fc - f * C;
    const int k  = threadIdx.x;                 // 0..255

#define FACE(g) (in + ((size_t)(g) * C + c) * fstride)

    float* O = out + (size_t)fc * HO * HO;

    float top, bot, lef, rig, tlv, trv, blv, brv;

    if (f < 4) {
        // ---- Northern polar face i = f ----
        const int i = f;
        const float* Ft  = FACE((i + 1) & 3);   // t  (also tr)
        const float* Ttl = FACE((i + 2) & 3);   // tl
        const float* Fl  = FACE((i + 3) & 3);   // l  (also bl)
        const float* Fb  = FACE(i + 4);         // b
        const float* Fbr = FACE(i + 8);         // br
        const float* Fr  = FACE(4 + ((i + 1) & 3)); // r
        top = Ft[(size_t)k * HP_W];             // rot(t,1)  -> t[k][0]
        bot = Fb[k];                            // b[0][k]
        lef = Fl[k];                            // rot(l,-1) -> l[0][k]
        rig = Fr[(size_t)k * HP_W];             // r[k][0]
        tlv = Ttl[0];                           // rot(tl,2) -> tl[0][0]
        trv = Ft[0];                            // rot(tr,1) -> tr[0][0]
        blv = Fl[HP_W - 1];                     // bl[0][255]
        brv = Fbr[0];                           // br[0][0]
    } else if (f < 8) {
        // ---- Equatorial face i = f-4 ----
        const int i = f - 4;
        const float* Ft  = FACE(i);                   // t
        const float* Fl  = FACE((i + 3) & 3);         // l
        const float* Fbl = FACE(4 + ((i + 3) & 3));   // bl
        const float* Fb  = FACE(8 + i);               // b
        const float* Fr  = FACE(8 + ((i + 3) & 3));   // r
        const float* Ftr = FACE(4 + ((i + 1) & 3));   // tr
        top = Ft[(size_t)(HP_H - 1) * HP_W + k];      // t[255][k]
        bot = Fb[k];                                  // b[0][k]
        lef = Fl[(size_t)k * HP_W + (HP_W - 1)];      // l[k][255]
        rig = Fr[(size_t)k * HP_W];                   // r[k][0]
        tlv = 0.5f * Ft[(size_t)(HP_H - 1) * HP_W] + 0.5f * Fl[HP_W - 1];
        trv = Ftr[(size_t)(HP_H - 1) * HP_W];         // tr[255][0]
        blv = Fbl[HP_W - 1];                          // bl[0][255]
        brv = 0.5f * Fb[HP_W - 1] + 0.5f * Fr[(size_t)(HP_H - 1) * HP_W];
    } else {
        // ---- Southern polar face i = f-8 ----
        const int i = f - 8;
        const float* Ft  = FACE(4 + ((i + 1) & 3));   // t
        const float* Ttl = FACE(i);                   // tl
        const float* Fl  = FACE(4 + i);               // l
        const float* Fbl = FACE(8 + ((i + 3) & 3));   // bl (also b)
        const float* Fbr = FACE(8 + ((i + 2) & 3));   // br
        const float* Fr  = FACE(8 + ((i + 1) & 3));   // r (also tr)
        const size_t LAST = (size_t)(HP_H - 1) * HP_W + (HP_W - 1);
        top = Ft[(size_t)(HP_H - 1) * HP_W + k];      // t[255][k]
        bot = Fbl[(size_t)k * HP_W + (HP_W - 1)];     // rot(b,1)  -> b[k][255]
        lef = Fl[(size_t)k * HP_W + (HP_W - 1)];      // l[k][255]
        rig = Fr[(size_t)(HP_H - 1) * HP_W + k];      // rot(r,-1) -> r[255][k]
        tlv = Ttl[LAST];                              // tl[255][255]
        trv = Fr[LAST];                               // rot(tr,-1)-> tr[255][255]
        blv = Fbl[LAST];                              // rot(bl,1) -> bl[255][255]
        brv = Fbr[LAST];                              // rot(br,2) -> br[255][255]
    }
#undef FACE

    O[1 + k]                              = top;  // row 0
    O[(size_t)(HO - 1) * HO + 1 + k]      = bot;  // row 257
    O[(size_t)(1 + k) * HO]               = lef;  // col 0
    O[(size_t)(1 + k) * HO + (HO - 1)]    = rig;  // col 257
    if (k == 0) {
        O[0]                                 = tlv;
        O[HO - 1]                            = trv;
        O[(size_t)(HO - 1) * HO]             = blv;
        O[(size_t)(HO - 1) * HO + (HO - 1)]  = brv;
    }
}

// ---------------------------------------------------------------------------
extern "C" void kernel_launch(void* const* d_in, const int* in_sizes, int n_in,
                              void* d_out, int out_size, void* d_ws, size_t ws_size,
                              hipStream_t stream) {
    (void)n_in; (void)d_ws; (void)ws_size;
    const float* x = (const float*)d_in[0];
    float* out = (float*)d_out;

    // Geometry: x is [1, 12, C, 256, 256]; derive C and p from the sizes
    // (the `pad` scalar lives in device memory; reading it would break capture).
    const int C = in_sizes[0] / (HP_F * HP_H * HP_W);             // 128
    const long long per = (long long)out_size / ((long long)HP_F * C); // HO*HO
    int HO = HP_H;
    while ((long long)HO * HO < per && HO < HP_H + 64) ++HO;      // 258
    const int p = (HO - HP_H) / 2;                                // 1

    if (p == 0) {
        hipMemcpyAsync(out, x, (size_t)out_size * sizeof(float),
                       hipMemcpyDeviceToDevice, stream);
        return;
    }

    // Bulk interior copy: 12*C*(256/8) blocks of 256 threads, async both ways.
    dim3 gridI(HP_F * C * (HP_H / HP_ROWS));
    healpix_interior_async<<<gridI, 256, 0, stream>>>(x, out, C, HO, p);

    // Halo (edges + corners), p==1 specialized.
    dim3 gridH(HP_F * C);
    healpix_halo_p1<<<gridH, 256, 0, stream>>>(x, out, C, HO);
}